// TypeGraphConvolution_72808285602376
// MI455X (gfx1250) — compile-verified
//
#include <hip/hip_runtime.h>

// Problem constants (match reference).
#define B_ 8
#define L_ 128
#define D_ 768
#define E_ 768

typedef float v2f __attribute__((ext_vector_type(2)));
typedef float v4f __attribute__((ext_vector_type(4)));
typedef float v8f __attribute__((ext_vector_type(8)));

// ---------------------------------------------------------------------------
// D = A(16x4 f32) * B(4x16 f32) + C(16x16 f32)   -> v_wmma_f32_16x16x4_f32
// 8-arg form: (neg_a, A, neg_b, B, c_mod, C, reuse_a, reuse_b)
// ---------------------------------------------------------------------------
static __device__ __forceinline__ v8f wmma_k4(v2f a, v2f b, v8f c) {
    return __builtin_amdgcn_wmma_f32_16x16x4_f32(false, a, false, b,
                                                 (short)0, c, false, false);
}

// ---------------------------------------------------------------------------
// Kernel 1 (bandwidth phase, ~402 MB streamed once):
//   S[b,j,e] = sum_i adj[b,j,i] * dep_embed[b,i,j,e]
// One block per (b,j). 192 threads, each owns 4 consecutive e's (float4).
// dep_embed is read exactly once -> non-temporal loads keep L2 clean for the
// weight matrices used by the WMMA kernels.
// ---------------------------------------------------------------------------
__global__ void __launch_bounds__(192)
k_reduce_dep(const float* __restrict__ adj,
             const float* __restrict__ dep,
             float* __restrict__ S) {
    const int bj = blockIdx.x;            // 0 .. B*L-1
    const int b  = bj / L_;
    const int j  = bj % L_;
    const int tid = threadIdx.x;          // 0 .. 191

    __shared__ float sadj[L_];
    if (tid < L_) sadj[tid] = adj[((size_t)b * L_ + j) * L_ + tid];
    __syncthreads();

    const size_t ebase = (size_t)4 * tid;
    // base at i = 0:  dep[(b*L*L + 0*L + j)*E + e]
    const float* p = dep + ((size_t)b * L_ * L_ + (size_t)j) * E_ + ebase;
    const size_t istride = (size_t)L_ * E_;   // stride between consecutive i

    v4f acc = {0.f, 0.f, 0.f, 0.f};
#pragma unroll 4
    for (int i = 0; i < L_; ++i) {
        v4f v = __builtin_nontemporal_load((const v4f*)(p + (size_t)i * istride));
        float w = sadj[i];
        acc.x += w * v.x;
        acc.y += w * v.y;
        acc.z += w * v.z;
        acc.w += w * v.w;
    }
    *(v4f*)(S + (size_t)bj * E_ + ebase) = acc;
}

// ---------------------------------------------------------------------------
// Kernel 2: P[b,j,d] = sum_i adj[b,j,i] * text[b,i,d]   (batched 128x128x768)
// One wave per 16x16 tile of P. EXEC all-ones (blockDim = 32, no divergence).
// ---------------------------------------------------------------------------
__global__ void __launch_bounds__(32)
k_adj_text(const float* __restrict__ adj,
           const float* __restrict__ text,
           float* __restrict__ P) {
    const int TN = D_ / 16;               // 48
    const int TM = L_ / 16;               // 8
    int t  = blockIdx.x;
    const int b  = t / (TM * TN);  t %= TM * TN;
    const int mt = t / TN;
    const int nt = t % TN;

    const int lane = threadIdx.x;
    const int half = lane >> 4;           // 0 | 1 -> K rows {0,1} vs {2,3}
    const int l16  = lane & 15;

    const int row = mt * 16 + l16;        // j
    const int col = nt * 16 + l16;        // d
    const float* Arow = adj  + ((size_t)b * L_ + row) * L_;   // A[row][i]
    const float* Bcol = text + (size_t)b * L_ * D_ + col;     // B[i][col]

    v8f c = {0.f, 0.f, 0.f, 0.f, 0.f, 0.f, 0.f, 0.f};
#pragma unroll 4
    for (int kk = 0; kk < L_; kk += 4) {
        const int k2 = kk + 2 * half;
        v2f a = *(const v2f*)(Arow + k2);            // contiguous in i
        v2f bb;
        bb.x = Bcol[(size_t)k2 * D_];                // strided over i
        bb.y = Bcol[(size_t)(k2 + 1) * D_];
        c = wmma_k4(a, bb, c);
    }

    float* Pt = P + ((size_t)b * L_ + (size_t)mt * 16) * D_ + (size_t)nt * 16;
#pragma unroll
    for (int r = 0; r < 8; ++r) {
        const int m = r + 8 * half;
        Pt[(size_t)m * D_ + l16] = c[r];
    }
}

// ---------------------------------------------------------------------------
// Kernel 3: P[n,d] += sum_e S[n,e] * dense_w[d,e]      (1024 x 768, K = 768)
// dense_w is [D,E] row-major -> B fragment is a contiguous float2 along e.
// Existing P is loaded as the C accumulator.
// ---------------------------------------------------------------------------
__global__ void __launch_bounds__(32)
k_s_densew(const float* __restrict__ S,
           const float* __restrict__ dw,
           float* __restrict__ P) {
    const int TN = D_ / 16;               // 48
    const int t  = blockIdx.x;
    const int mt = t / TN;                // 0..63 row tiles of 1024 rows
    const int nt = t % TN;

    const int lane = threadIdx.x;
    const int half = lane >> 4;
    const int l16  = lane & 15;

    const int row = mt * 16 + l16;        // n = b*L + j
    const int col = nt * 16 + l16;        // d
    const float* Arow = S  + (size_t)row * E_;
    const float* Brow = dw + (size_t)col * E_;       // B[e][col] = dw[col*E+e]

    float* Pt = P + (size_t)mt * 16 * D_ + (size_t)nt * 16;
    v8f c;
#pragma unroll
    for (int r = 0; r < 8; ++r) {
        const int m = r + 8 * half;
        c[r] = Pt[(size_t)m * D_ + l16];
    }

#pragma unroll 4
    for (int kk = 0; kk < E_; kk += 4) {
        const int k2 = kk + 2 * half;
        v2f a  = *(const v2f*)(Arow + k2);
        v2f bb = *(const v2f*)(Brow + k2);           // contiguous float2
        c = wmma_k4(a, bb, c);
    }

#pragma unroll
    for (int r = 0; r < 8; ++r) {
        const int m = r + 8 * half;
        Pt[(size_t)m * D_ + l16] = c[r];
    }
}

// ---------------------------------------------------------------------------
// Kernel 4: out[n,f] = sum_d P[n,d] * weight[d,f] + bias[f]
// ---------------------------------------------------------------------------
__global__ void __launch_bounds__(32)
k_final(const float* __restrict__ P,
        const float* __restrict__ W,
        const float* __restrict__ bias,
        float* __restrict__ out) {
    const int TN = D_ / 16;
    const int t  = blockIdx.x;
    const int mt = t / TN;
    const int nt = t % TN;

    const int lane = threadIdx.x;
    const int half = lane >> 4;
    const int l16  = lane & 15;

    const int row = mt * 16 + l16;        // n
    const int col = nt * 16 + l16;        // f
    const float* Arow = P + (size_t)row * D_;
    const float* Bcol = W + col;          // W[d*D + col]

    v8f c = {0.f, 0.f, 0.f, 0.f, 0.f, 0.f, 0.f, 0.f};
#pragma unroll 4
    for (int kk = 0; kk < D_; kk += 4) {
        const int k2 = kk + 2 * half;
        v2f a = *(const v2f*)(Arow + k2);
        v2f bb;
        bb.x = Bcol[(size_t)k2 * D_];
        bb.y = Bcol[(size_t)(k2 + 1) * D_];
        c = wmma_k4(a, bb, c);
    }

    const float bv = bias[col];           // per-column bias (column = lane&15)
    float* Ot = out + (size_t)mt * 16 * D_ + (size_t)nt * 16;
#pragma unroll
    for (int r = 0; r < 8; ++r) {
        const int m = r + 8 * half;
        Ot[(size_t)m * D_ + l16] = c[r] + bv;
    }
}

// ---------------------------------------------------------------------------
extern "C" void kernel_launch(void* const* d_in, const int* in_sizes, int n_in,
                              void* d_out, int out_size, void* d_ws, size_t ws_size,
                              hipStream_t stream) {
    const float* text    = (const float*)d_in[0];   // [B,L,D]
    const float* adj     = (const float*)d_in[1];   // [B,L,L]
    const float* dep     = (const float*)d_in[2];   // [B,L,L,E]
    const float* weight  = (const float*)d_in[3];   // [D,D]
    const float* dense_w = (const float*)d_in[4];   // [D,E]
    const float* bias    = (const float*)d_in[5];   // [D]
    float* out = (float*)d_out;                     // [B,L,D]

    float* S = (float*)d_ws;                        // [B*L, E]  (3 MB)
    float* P = S + (size_t)B_ * L_ * E_;            // [B*L, D]  (3 MB)

    const int TN = D_ / 16;                         // 48
    const int TM = L_ / 16;                         // 8
    const int rows_tiles = (B_ * L_) / 16;          // 64

    // Phase 1: stream dep_embed once (memory-bound floor, ~402 MB).
    k_reduce_dep<<<B_ * L_, 192, 0, stream>>>(adj, dep, S);

    // Phase 2a: P = adj @ text (batched).
    k_adj_text<<<B_ * TM * TN, 32, 0, stream>>>(adj, text, P);

    // Phase 2b: P += S @ dense_w^T.
    k_s_densew<<<rows_tiles * TN, 32, 0, stream>>>(S, dense_w, P);

    // Phase 3: out = P @ weight + bias.
    k_final<<<rows_tiles * TN, 32, 0, stream>>>(P, weight, bias, out);
}